// DeformableTransformer_53472342835864
// MI455X (gfx1250) — compile-verified
//
#include <hip/hip_runtime.h>
#include <hip/hip_bf16.h>

// ---------------------------------------------------------------------------
// CDNA5 (gfx1250) deformable-transformer forward.
// GEMMs: V_WMMA_F32_16X16X4_F32, 128x64 tiles, async global->LDS staging
// (GLOBAL_LOAD_ASYNC_TO_LDS_B128 + s_wait_asynccnt) when available.
// Workspace requirement: ~460 MB.
// ---------------------------------------------------------------------------

typedef float v2f __attribute__((ext_vector_type(2)));
typedef float v8f __attribute__((ext_vector_type(8)));
typedef int   vi4 __attribute__((vector_size(16)));

#define B_    2
#define S_    15795
#define C_    256
#define HH_   8
#define DH_   32
#define L_    4
#define P_    4
#define NQ_   80
#define FF_   1024
#define NENC_ 6
#define NDEC_ 6

#define LDSPAD 20   // floats per LDS row (80B: 16B-aligned rows, conflict-free)

#if defined(__HIP_DEVICE_COMPILE__) && __has_builtin(__builtin_amdgcn_global_load_async_to_lds_b128)
#define ASYNC_LDS 1
#endif

#ifdef ASYNC_LDS
static __device__ __forceinline__ void async_ld128(const float* g, float* l)
{
  // global -> LDS, 16 bytes, tracked by ASYNCcnt
  __builtin_amdgcn_global_load_async_to_lds_b128((vi4*)g, (vi4*)l, 0, 0);
}
#endif

static __device__ __forceinline__ void wait_async_lds()
{
#ifdef ASYNC_LDS
  asm volatile("s_wait_asynccnt 0x0" ::: "memory");
#endif
}

// ============================================================
// GEMM: Y[M,N] = X[M,K] @ W[N,K]^T + bias[N]  (optional ReLU)
// Block tile 128(M) x 64(N), BK=16. 8 waves; wave w computes rows
// [w*16, w*16+16) x all 64 cols via 4 f32-WMMA accumulators.
// Requires: K % 16 == 0, N % 64 == 0 (true for all call sites).
// ============================================================
__global__ __launch_bounds__(256) void gemm_bias(const float* __restrict__ X,
                                                 const float* __restrict__ Wt,
                                                 const float* __restrict__ bias,
                                                 float* __restrict__ Y,
                                                 int M, int N, int K, int relu)
{
  __shared__ float sA[128][LDSPAD];   // A tile, row-major
  __shared__ float sB[64][LDSPAD];    // B tile, row-major: sB[n][k] = W[n0+n][k0+k]
  const int tid   = threadIdx.x;
  const int lane  = tid & 31;
  const int wid   = tid >> 5;          // 0..7 -> 16-row strip
  const int lrow  = lane & 15;
  const int khalf = (lane >> 4) << 1;  // 0 or 2 (K-half per ISA A/B layout)
  const int m0 = blockIdx.y * 128;
  const int n0 = blockIdx.x * 64;
  const bool mfull = (m0 + 128 <= M);

  v8f acc0 = {}, acc1 = {}, acc2 = {}, acc3 = {};

  for (int k0 = 0; k0 < K; k0 += 16) {
#ifdef ASYNC_LDS
    if (mfull) {
      // A tile: 128 rows x 16 cols = 512 16B-chunks, 2 per thread
      #pragma unroll
      for (int i = 0; i < 2; i++) {
        int ch = tid + i * 256;
        int r = ch >> 2, cq = (ch & 3) << 2;
        async_ld128(X + (size_t)(m0 + r) * K + k0 + cq, &sA[r][cq]);
      }
    } else
#endif
    {
      #pragma unroll
      for (int i = 0; i < 8; i++) {
        int e = tid + i * 256;
        int r = e >> 4, c = e & 15;
        int gm = m0 + r;
        const float* ap = X + (size_t)gm * K + k0 + c;
        sA[r][c] = (gm < M) ? ap[0] : 0.f;
        __builtin_prefetch(ap + 16, 0, 1);  // speculative: OOB silently dropped
      }
    }
#ifdef ASYNC_LDS
    {
      // B tile: 64 rows x 16 cols = 256 16B-chunks, 1 per thread
      int r = tid >> 2, cq = (tid & 3) << 2;
      async_ld128(Wt + (size_t)(n0 + r) * K + k0 + cq, &sB[r][cq]);
    }
#else
    #pragma unroll
    for (int i = 0; i < 4; i++) {
      int e = tid + i * 256;
      int n = e >> 4, k = e & 15;
      const float* bp = Wt + (size_t)(n0 + n) * K + k0 + k;
      sB[n][k] = bp[0];
      __builtin_prefetch(bp + 16, 0, 1);
    }
#endif
    wait_async_lds();
    __syncthreads();

    #pragma unroll
    for (int kk = 0; kk < 4; kk++) {
      const int kb = kk * 4;
      const float* ar = &sA[wid * 16 + lrow][kb + khalf];
      v2f a; a.x = ar[0]; a.y = ar[1];     // contiguous -> ds_load_b64
      const float* b0r = &sB[lrow][kb + khalf];
      const float* b1r = &sB[16 + lrow][kb + khalf];
      const float* b2r = &sB[32 + lrow][kb + khalf];
      const float* b3r = &sB[48 + lrow][kb + khalf];
      v2f b0; b0.x = b0r[0]; b0.y = b0r[1];
      v2f b1; b1.x = b1r[0]; b1.y = b1r[1];
      v2f b2; b2.x = b2r[0]; b2.y = b2r[1];
      v2f b3; b3.x = b3r[0]; b3.y = b3r[1];
      acc0 = __builtin_amdgcn_wmma_f32_16x16x4_f32(false, a, false, b0, (short)0, acc0, false, false);
      acc1 = __builtin_amdgcn_wmma_f32_16x16x4_f32(false, a, false, b1, (short)0, acc1, false, false);
      acc2 = __builtin_amdgcn_wmma_f32_16x16x4_f32(false, a, false, b2, (short)0, acc2, false, false);
      acc3 = __builtin_amdgcn_wmma_f32_16x16x4_f32(false, a, false, b3, (short)0, acc3, false, false);
    }
    __syncthreads();
  }

  // D layout: VGPR v -> M = v + 8*(lane>=16), N = lane%16
  const int mrow0 = m0 + wid * 16 + ((lane >> 4) << 3);
  #pragma unroll
  for (int v = 0; v < 8; v++) {
    int m = mrow0 + v;
    if (m < M) {
      #pragma unroll
      for (int t = 0; t < 4; t++) {
        int n = n0 + t * 16 + lrow;
        float y = (t == 0 ? acc0[v] : t == 1 ? acc1[v] : t == 2 ? acc2[v] : acc3[v]) + bias[n];
        if (relu) y = fmaxf(y, 0.f);
        Y[(size_t)m * N + n] = y;
      }
    }
  }
}

// ============================================================
// elementwise add
// ============================================================
__global__ void add_kernel(const float* __restrict__ a, const float* __restrict__ b,
                           float* __restrict__ y, size_t n)
{
  size_t i = (size_t)blockIdx.x * blockDim.x + threadIdx.x;
  size_t stride = (size_t)gridDim.x * blockDim.x;
  for (; i < n; i += stride) y[i] = a[i] + b[i];
}

// ============================================================
// LayerNorm over C=256: Y = LN(R + D) * g + b   (in-place safe)
// ============================================================
__global__ __launch_bounds__(256) void ln_kernel(const float* __restrict__ R,
                                                 const float* __restrict__ Dl,
                                                 const float* __restrict__ g,
                                                 const float* __restrict__ bb,
                                                 float* __restrict__ Y)
{
  const int row = blockIdx.x, c = threadIdx.x;
  const size_t base = (size_t)row * C_;
  float x = R[base + c] + Dl[base + c];
  __shared__ float red[256];
  red[c] = x; __syncthreads();
  for (int s = 128; s > 0; s >>= 1) { if (c < s) red[c] += red[c + s]; __syncthreads(); }
  float mu = red[0] * (1.f / C_); __syncthreads();
  float d = x - mu;
  red[c] = d * d; __syncthreads();
  for (int s = 128; s > 0; s >>= 1) { if (c < s) red[c] += red[c + s]; __syncthreads(); }
  float var = red[0] * (1.f / C_);
  Y[base + c] = d * rsqrtf(var + 1e-5f) * g[c] + bb[c];
}

// ============================================================
// 3D multi-scale deformable sampling.
// One wave per (b, q, h); lane = channel d (DH_ == 32 == wave32).
// ============================================================
__global__ __launch_bounds__(256) void ms_sample(const float* __restrict__ val,
                                                 const float* __restrict__ off,
                                                 const float* __restrict__ awl,
                                                 const float* __restrict__ vr,
                                                 float* __restrict__ out)
{
  const int gw = blockIdx.x * 8 + (threadIdx.x >> 5);
  const int lane = threadIdx.x & 31;
  if (gw >= B_ * S_ * HH_) return;
  const int h = gw % HH_;
  const int q = (gw / HH_) % S_;
  const int b = gw / (HH_ * S_);

  const int lD[4]  = {24, 12, 6, 3};
  const int lst[4] = {0, 13824, 15552, 15768};

  int lt = (q < 13824) ? 0 : (q < 15552) ? 1 : (q < 15768) ? 2 : 3;
  int dim = lD[lt];
  int idx = q - lst[lt];
  float rx = ((idx % dim) + 0.5f) / dim;
  float ry = (((idx / dim) % dim) + 0.5f) / dim;
  float rz = ((idx / (dim * dim)) + 0.5f) / dim;

  const float* aw = awl + ((size_t)(b * S_ + q)) * (HH_ * L_ * P_) + h * (L_ * P_);
  float wv_[16]; float mx = -1e30f;
  #pragma unroll
  for (int i = 0; i < 16; i++) { wv_[i] = aw[i]; mx = fmaxf(mx, wv_[i]); }
  float sum = 0.f;
  #pragma unroll
  for (int i = 0; i < 16; i++) { wv_[i] = __expf(wv_[i] - mx); sum += wv_[i]; }
  const float inv = 1.f / sum;

  const float* offp = off + ((size_t)(b * S_ + q)) * (HH_ * L_ * P_ * 3) + h * (L_ * P_ * 3);
  float acc = 0.f;
  for (int l = 0; l < L_; l++) {
    const int dl = lD[l];
    const int s0 = lst[l];
    const float fd = (float)dl;
    const float vrx = vr[(b * L_ + l) * 3 + 0];
    const float vry = vr[(b * L_ + l) * 3 + 1];
    const float vrz = vr[(b * L_ + l) * 3 + 2];
    const float bx = rx * vrx, by = ry * vry, bz = rz * vrz;
    for (int p = 0; p < P_; p++) {
      const float* o3 = offp + l * (P_ * 3) + p * 3;
      float x = (bx + o3[0] / fd) * fd - 0.5f;
      float y = (by + o3[1] / fd) * fd - 0.5f;
      float z = (bz + o3[2] / fd) * fd - 0.5f;
      float x0 = floorf(x), y0 = floorf(y), z0 = floorf(z);
      float aww = wv_[l * 4 + p] * inv;
      float s = 0.f;
      #pragma unroll
      for (int dz = 0; dz < 2; dz++)
        #pragma unroll
        for (int dy = 0; dy < 2; dy++)
          #pragma unroll
          for (int dx = 0; dx < 2; dx++) {
            float xi = x0 + dx, yi = y0 + dy, zi = z0 + dz;
            float w = (1.f - fabsf(x - xi)) * (1.f - fabsf(y - yi)) * (1.f - fabsf(z - zi));
            bool ok = (xi >= 0.f) && (xi < fd) && (yi >= 0.f) && (yi < fd) &&
                      (zi >= 0.f) && (zi < fd);
            int xc = min(max((int)xi, 0), dl - 1);
            int yc = min(max((int)yi, 0), dl - 1);
            int zc = min(max((int)zi, 0), dl - 1);
            int vi = (zc * dl + yc) * dl + xc;
            float sv = val[((size_t)(b * S_) + s0 + vi) * C_ + h * DH_ + lane];
            s += ok ? w * sv : 0.f;
          }
      acc += aww * s;
    }
  }
  out[((size_t)(b * S_ + q)) * C_ + h * DH_ + lane] = acc;
}

// ============================================================
// Decoder self-attention (80 queries x 80 keys) per (b,h) block.
// ============================================================
__global__ __launch_bounds__(256) void sa_attn(const float* __restrict__ QK, // 160 x 512
                                               const float* __restrict__ Vv, // 160 x 256
                                               float* __restrict__ O)        // 160 x 256
{
  const int b = blockIdx.x / HH_;
  const int h = blockIdx.x % HH_;
  __shared__ float lg[NQ_][NQ_];
  const float scale = 0.17677669529663687f; // 1/sqrt(32)
  for (int e = threadIdx.x; e < NQ_ * NQ_; e += 256) {
    int qi = e / NQ_, ki = e % NQ_;
    const float* qp = QK + (size_t)(b * NQ_ + qi) * 512 + h * DH_;
    const float* kp = QK + (size_t)(b * NQ_ + ki) * 512 + 256 + h * DH_;
    float d = 0.f;
    #pragma unroll
    for (int j = 0; j < DH_; j++) d += qp[j] * kp[j];
    lg[qi][ki] = d * scale;
  }
  __syncthreads();
  if (threadIdx.x < NQ_) {
    int qi = threadIdx.x;
    float mx = -1e30f;
    for (int k = 0; k < NQ_; k++) mx = fmaxf(mx, lg[qi][k]);
    float sum = 0.f;
    for (int k = 0; k < NQ_; k++) { float e = __expf(lg[qi][k] - mx); lg[qi][k] = e; sum += e; }
    float inv = 1.f / sum;
    for (int k = 0; k < NQ_; k++) lg[qi][k] *= inv;
  }
  __syncthreads();
  for (int e = threadIdx.x; e < NQ_ * DH_; e += 256) {
    int qi = e / DH_, d = e % DH_;
    float o = 0.f;
    for (int k = 0; k < NQ_; k++) o += lg[qi][k] * Vv[(size_t)(b * NQ_ + k) * C_ + h * DH_ + d];
    O[(size_t)(b * NQ_ + qi) * C_ + h * DH_ + d] = o;
  }
}

// ============================================================
// Cross-attention: logits (masked), row softmax, weighted sum.
// ============================================================
__global__ __launch_bounds__(256) void ca_logits(const float* __restrict__ Q,  // 160 x 256
                                                 const float* __restrict__ KH, // B*S x 256
                                                 const unsigned char* __restrict__ mask, // NQ x S
                                                 float* __restrict__ Lg)
{
  const int gw = blockIdx.x * 8 + (threadIdx.x >> 5);
  const int lane = threadIdx.x & 31;
  const int qi = gw % NQ_;
  const int h = (gw / NQ_) % HH_;
  const int b = gw / (NQ_ * HH_);
  const float qv = Q[(size_t)(b * NQ_ + qi) * C_ + h * DH_ + lane];
  const float scale = 0.17677669529663687f;
  const size_t row = (size_t)gw * S_;
  for (int k = lane; k < S_; k += 32) {
    const float* kp = KH + ((size_t)(b * S_) + k) * C_ + h * DH_;
    float d = 0.f;
    #pragma unroll
    for (int j = 0; j < DH_; j++) d += __shfl(qv, j, 32) * kp[j];
    d *= scale;
    if (mask[(size_t)qi * S_ + k]) d = -1e9f;
    Lg[row + k] = d;
  }
}

__global__ __launch_bounds__(256) void row_softmax(float* __restrict__ Lg, int len)
{
  const size_t base = (size_t)blockIdx.x * len;
  __shared__ float red[256];
  const int t = threadIdx.x;
  float mx = -1e30f;
  for (int i = t; i < len; i += 256) mx = fmaxf(mx, Lg[base + i]);
  red[t] = mx; __syncthreads();
  for (int s = 128; s > 0; s >>= 1) { if (t < s) red[t] = fmaxf(red[t], red[t + s]); __syncthreads(); }
  mx = red[0]; __syncthreads();
  float sum = 0.f;
  for (int i = t; i < len; i += 256) { float e = __expf(Lg[base + i] - mx); Lg[base + i] = e; sum += e; }
  red[t] = sum; __syncthreads();
  for (int s = 128; s > 0; s >>= 1) { if (t < s) red[t] += red[t + s]; __syncthreads(); }
  float inv = 1.f / red[0];
  __syncthreads();
  for (int i = t; i < len; i += 256) Lg[base + i] *= inv;
}

__global__ __launch_bounds__(256) void ca_out(const float* __restrict__ Lg,
                                              const float* __restrict__ VH, // B*S x 256
                                              float* __restrict__ O)        // 160 x 256
{
  const int gw = blockIdx.x * 8 + (threadIdx.x >> 5);
  const int lane = threadIdx.x & 31;
  const int qi = gw % NQ_;
  const int h = (gw / NQ_) % HH_;
  const int b = gw / (NQ_ * HH_);
  const size_t row = (size_t)gw * S_;
  const float* vp = VH + (size_t)(b * S_) * C_ + h * DH_ + lane;
  float acc = 0.f;
  for (int k = 0; k < S_; k++) acc += Lg[row + k] * vp[(size_t)k * C_];
  O[(size_t)(b * NQ_ + qi) * C_ + h * DH_ + lane] = acc;
}

// ============================================================
// host orchestration
// ============================================================
extern "C" void kernel_launch(void* const* d_in, const int* in_sizes, int n_in,
                              void* d_out, int out_size, void* d_ws, size_t ws_size,
                              hipStream_t stream)
{
  const float* src        = (const float*)d_in[0];
  const float* pos        = (const float*)d_in[1];
  const float* lvl_pos    = (const float*)d_in[2];
  const float* tgt        = (const float*)d_in[3];
  const float* query_pos  = (const float*)d_in[4];
  const float* vratios    = (const float*)d_in[5];
  const float* eow = (const float*)d_in[6];   const float* eob = (const float*)d_in[7];
  const float* eaw = (const float*)d_in[8];   const float* eab = (const float*)d_in[9];
  const float* evw = (const float*)d_in[10];  const float* evb = (const float*)d_in[11];
  const float* eouw = (const float*)d_in[12]; const float* eoub = (const float*)d_in[13];
  const float* el1g = (const float*)d_in[14]; const float* el1b = (const float*)d_in[15];
  const float* ef1w = (const float*)d_in[16]; const float* ef1b = (const float*)d_in[17];
  const float* ef2w = (const float*)d_in[18]; const float* ef2b = (const float*)d_in[19];
  const float* el2g = (const float*)d_in[20]; const float* el2b = (const float*)d_in[21];
  const float* dsiw = (const float*)d_in[22]; const float* dsib = (const float*)d_in[23];
  const float* dsow = (const float*)d_in[24]; const float* dsob = (const float*)d_in[25];
  const float* dciw = (const float*)d_in[26]; const float* dcib = (const float*)d_in[27];
  const float* dcow = (const float*)d_in[28]; const float* dcob = (const float*)d_in[29];
  const float* dl1g = (const float*)d_in[30]; const float* dl1b = (const float*)d_in[31];
  const float* dl2g = (const float*)d_in[32]; const float* dl2b = (const float*)d_in[33];
  const float* dl3g = (const float*)d_in[34]; const float* dl3b = (const float*)d_in[35];
  const float* df1w = (const float*)d_in[36]; const float* df1b = (const float*)d_in[37];
  const float* df2w = (const float*)d_in[38]; const float* df2b = (const float*)d_in[39];
  const unsigned char* attn_mask = (const unsigned char*)d_in[40]; // jax bool -> byte
  (void)d_in; (void)in_sizes; (void)n_in; (void)out_size; (void)ws_size;

  float* ws = (float*)d_ws;
  size_t cur = 0;
  auto alloc = [&](size_t n) { float* p = ws + cur; cur += (n + 63) & ~(size_t)63; return p; };

  const size_t MS = (size_t)B_ * S_;   // 31590
  const int Menc = (int)MS;
  const size_t MQ = (size_t)B_ * NQ_;  // 160
  const int Mq = (int)MQ;

  float* x    = alloc(MS * C_);      // encoder state, becomes `memory`
  float* xin  = alloc(MS * C_);
  float* val  = alloc(MS * C_);
  float* samp = alloc(MS * C_);
  float* tmp  = alloc(MS * C_);
  float* offb = alloc(MS * 384);
  float* awb  = alloc(MS * 128);
  float* hbuf = alloc(MS * FF_);     // encoder FFN hidden; aliased as CA logits in decoder
  float* memlp = alloc(MS * C_);
  float* khb  = alloc(MS * C_);
  float* vhb  = alloc(MS * C_);
  float* qbuf = alloc(MQ * C_);
  float* qk   = alloc(MQ * C_);
  float* saqk = alloc(MQ * 512);
  float* sav  = alloc(MQ * C_);
  float* sao  = alloc(MQ * C_);
  float* dtmp = alloc(MQ * C_);
  float* dh   = alloc(MQ * FF_);
  float* caq  = alloc(MQ * C_);
  float* cal  = hbuf;  // 1280*S_ floats fits in MS*FF_ region

  auto gemm = [&](const float* X, const float* W, const float* bias, float* Y,
                  int M, int N, int K, int relu) {
    dim3 g((N + 63) / 64, (M + 127) / 128);
    gemm_bias<<<g, 256, 0, stream>>>(X, W, bias, Y, M, N, K, relu);
  };
  auto addv = [&](const float* a, const float* b, float* y, size_t n) {
    add_kernel<<<4096, 256, 0, stream>>>(a, b, y, n);
  };
  auto ln = [&](const float* R, const float* Dl, const float* g, const float* b,
                float* Y, int rows) {
    ln_kernel<<<rows, 256, 0, stream>>>(R, Dl, g, b, Y);
  };

  // ---- encoder ----
  (void)hipMemcpyAsync(x, src, MS * C_ * sizeof(float), hipMemcpyDeviceToDevice, stream);
  for (int i = 0; i < NENC_; i++) {
    const float* eow_i = eow + (size_t)i * 384 * C_;  const float* eob_i = eob + (size_t)i * 384;
    const float* eaw_i = eaw + (size_t)i * 128 * C_;  const float* eab_i = eab + (size_t)i * 128;
    const float* evw_i = evw + (size_t)i * C_ * C_;   const float* evb_i = evb + (size_t)i * C_;
    const float* eouw_i = eouw + (size_t)i * C_ * C_; const float* eoub_i = eoub + (size_t)i * C_;
    const float* ef1w_i = ef1w + (size_t)i * FF_ * C_; const float* ef1b_i = ef1b + (size_t)i * FF_;
    const float* ef2w_i = ef2w + (size_t)i * C_ * FF_; const float* ef2b_i = ef2b + (size_t)i * C_;

    addv(x, pos, xin, MS * C_);                               // query = out + pos
    gemm(x,   evw_i, evb_i, val,  Menc, C_,  C_, 0);          // value proj (on out)
    gemm(xin, eow_i, eob_i, offb, Menc, 384, C_, 0);          // sampling offsets
    gemm(xin, eaw_i, eab_i, awb,  Menc, 128, C_, 0);          // attention logits
    ms_sample<<<(B_ * S_ * HH_) / 8, 256, 0, stream>>>(val, offb, awb, vratios, samp);
    gemm(samp, eouw_i, eoub_i, tmp, Menc, C_, C_, 0);         // output proj
    ln(x, tmp, el1g + (size_t)i * C_, el1b + (size_t)i * C_, x, Menc);
    gemm(x, ef1w_i, ef1b_i, hbuf, Menc, FF_, C_, 1);          // fc1 + relu
    gemm(hbuf, ef2w_i, ef2b_i, tmp, Menc, C_, FF_, 0);        // fc2
    ln(x, tmp, el2g + (size_t)i * C_, el2b + (size_t)i * C_, x, Menc);
  }
  // memory = x
  addv(x, lvl_pos, memlp, MS * C_);

  // ---- decoder ----
  (void)hipMemcpyAsync(qbuf, tgt, MQ * C_ * sizeof(float), hipMemcpyDeviceToDevice, stream);
  for (int i = 0; i < NDEC_; i++) {
    const float* dsiw_i = dsiw + (size_t)i * 3 * C_ * C_; const float* dsib_i = dsib + (size_t)i * 3 * C_;
    const float* dsow_i = dsow + (size_t)i * C_ * C_;     const float* dsob_i = dsob + (size_t)i * C_;
    const float* dciw_i = dciw + (size_t)i * 3 * C_ * C_; const float* dcib_i = dcib + (size_t)i * 3 * C_;
    const float* dcow_i = dcow + (size_t)i * C_ * C_;     const float* dcob_i = dcob + (size_t)i * C_;
    const float* df1w_i = df1w + (size_t)i * FF_ * C_;    const float* df1b_i = df1b + (size_t)i * FF_;
    const float* df2w_i = df2w + (size_t)i * C_ * FF_;    const float* df2b_i = df2b + (size_t)i * C_;

    // self-attention: q,k from q+query_pos ; v from q
    addv(qbuf, query_pos, qk, MQ * C_);
    gemm(qk,   dsiw_i,            dsib_i,            saqk, Mq, 512, C_, 0);
    gemm(qbuf, dsiw_i + 512 * C_, dsib_i + 512,      sav,  Mq, C_,  C_, 0);
    sa_attn<<<B_ * HH_, 256, 0, stream>>>(saqk, sav, sao);
    gemm(sao, dsow_i, dsob_i, dtmp, Mq, C_, C_, 0);
    ln(qbuf, dtmp, dl2g + (size_t)i * C_, dl2b + (size_t)i * C_, qbuf, Mq);

    // cross-attention: q from q+query_pos; k from memory+lvl_pos; v from memory
    addv(qbuf, query_pos, qk, MQ * C_);
    gemm(qk,    dciw_i,            dcib_i,       caq, Mq,   C_, C_, 0);
    gemm(memlp, dciw_i + 256 * C_, dcib_i + 256, khb, Menc, C_, C_, 0);
    gemm(x,     dciw_i + 512 * C_, dcib_i + 512, vhb, Menc, C_, C_, 0);
    ca_logits<<<(B_ * HH_ * NQ_) / 8, 256, 0, stream>>>(caq, khb, attn_mask, cal);
    row_softmax<<<B_ * HH_ * NQ_, 256, 0, stream>>>(cal, S_);
    ca_out<<<(B_ * HH_ * NQ_) / 8, 256, 0, stream>>>(cal, vhb, sao);
    gemm(sao, dcow_i, dcob_i, dtmp, Mq, C_, C_, 0);
    ln(qbuf, dtmp, dl1g + (size_t)i * C_, dl1b + (size_t)i * C_, qbuf, Mq);

    // FFN
    gemm(qbuf, df1w_i, df1b_i, dh, Mq, FF_, C_, 1);
    gemm(dh, df2w_i, df2b_i, dtmp, Mq, C_, FF_, 0);
    float* yout = (i == NDEC_ - 1) ? (float*)d_out : qbuf;
    ln(qbuf, dtmp, dl3g + (size_t)i * C_, dl3b + (size_t)i * C_, yout, Mq);
  }
}